// EMD_33646773796928
// MI455X (gfx1250) — compile-verified
//
#include <hip/hip_runtime.h>

// CDNA5 / gfx1250 Sinkhorn-EMD.
// Cost tiles via V_WMMA_F32_16X16X4_F32; per-batch work split across 8
// workgroups synchronized by a sense-reversing global barrier so the
// trans-pipe-bound exp work spreads over ~128 WGPs instead of 16.

typedef __attribute__((ext_vector_type(2))) float v2f;
typedef __attribute__((ext_vector_type(8))) float v8f;

#define BATCH   16
#define NPT     2048
#define NT      (NPT / 16)          // 128 tiles per dimension
#define ITERS   100

#define EPS_F   0.05f
#define LOG2E   1.4426950408889634f
#define LN2     0.6931471805599453f
#define KC      (LOG2E / EPS_F)
#define LN_N    7.6246189861593985f // ln(2048)

// ---- multi-WG configuration ----
#define NWGB    8                   // workgroups per batch
#define NWG     (BATCH * NWGB)      // 128 workgroups
#define WAVES   8                   // waves per workgroup
#define TPB     (WAVES * 32)        // 256 threads
#define BWAVES  (NWGB * WAVES)      // 64 waves cooperating on one batch
#define SPW     (NT / BWAVES)       // 2 row-strips per wave per pass

// ---- workspace layout (element offsets) ----
#define F_OFF   0
#define G_OFF   (BATCH * NPT)
#define P_OFF   (2 * BATCH * NPT)               // NWG partial sums
#define BAR_OFF (P_OFF + NWG)                   // BATCH * 64 uints (padded)
#define WS_FLOATS (BAR_OFF + BATCH * 64)
#define WS_BYTES_NEEDED ((size_t)WS_FLOATS * 4)

// =======================================================================
// shared operand storage
// =======================================================================
struct OperandBuf {
    // B-operand formatted arrays: lanes 0-15 read Lo=(c0,c1); 16-31 Hi=(c2,1)
    v2f   bXlo[NPT];
    v2f   bXhi[NPT];
    v2f   bYlo[NPT];
    v2f   bYhi[NPT];
    float xn[NPT];
    float yn[NPT];
    float aux[NPT];
    float wsum[32];
};

struct OperandBufFB {                 // fallback: potentials live in LDS too
    v2f   bXlo[NPT];
    v2f   bXhi[NPT];
    v2f   bYlo[NPT];
    v2f   bYhi[NPT];
    float xn[NPT];
    float yn[NPT];
    float f[NPT];
    float g[NPT];
    float aux[NPT];
    float wsum[32];
};

// =======================================================================
// one logsumexp pass over NT column tiles for this wave's strips.
// outPot may be LDS or global; strips = waveId + s*strideWaves.
// =======================================================================
__device__ __forceinline__ void lse_pass(
    const v2f* __restrict__ rowLo, const v2f* __restrict__ rowHi,
    const float* __restrict__ rowN,
    const v2f* __restrict__ colLo, const v2f* __restrict__ colHi,
    const float* __restrict__ gle,
    float* __restrict__ outPot,
    int waveId, int strideWaves, int nStrips, int lane)
{
    const int half = lane >> 4;
    const int l    = lane & 15;
    const v2f* col = half ? colHi : colLo;

    for (int sidx = 0; sidx < nStrips; ++sidx) {
        const int strip = (waveId + sidx * strideWaves) * 16;

        // A operand, 16x4 f32 layout (lanes 0-15: K0,K1; 16-31: K2,K3)
        // row m: (-2r0, -2r1, -2r2, ||r||^2)
        v2f a;
        if (half == 0) {
            v2f t = rowLo[strip + l];
            a.x = -2.0f * t.x; a.y = -2.0f * t.y;
        } else {
            v2f t = rowHi[strip + l];
            a.x = -2.0f * t.x; a.y = rowN[strip + l];
        }

        float acc[8];
#pragma unroll
        for (int k = 0; k < 8; ++k) acc[k] = 0.0f;
        v8f cz = {};

#pragma unroll 2
        for (int jt = 0; jt < NT; ++jt) {
            const int n  = jt * 16 + l;
            v2f   bb = col[n];
            float gl = gle[n];
            v8f d = __builtin_amdgcn_wmma_f32_16x16x4_f32(
                false, a, false, bb, (short)0, cz, false, false);
#pragma unroll
            for (int k = 0; k < 8; ++k) {
                float t2 = __builtin_fmaf(d[k], -KC, gl);
                acc[k] += __builtin_amdgcn_exp2f(t2);
            }
        }

        // fixed-order tree reduction over the 16 columns in each half-wave
#pragma unroll
        for (int k = 0; k < 8; ++k) {
#pragma unroll
            for (int m = 1; m < 16; m <<= 1)
                acc[k] += __shfl_xor(acc[k], m, 32);
        }
        if (l == 0) {
            const int base = strip + half * 8;
            float r[8];
#pragma unroll
            for (int k = 0; k < 8; ++k)
                r[k] = -EPS_F * (LN_N + LN2 * __builtin_amdgcn_logf(acc[k]));
            float4* o = reinterpret_cast<float4*>(outPot + base);
            o[0] = make_float4(r[0], r[1], r[2], r[3]);
            o[1] = make_float4(r[4], r[5], r[6], r[7]);
        }
    }
}

// final pass: partial sum of P_ij * C_ij over this wave's strips
__device__ __forceinline__ float dis_pass(
    const v2f* __restrict__ rowLo, const v2f* __restrict__ rowHi,
    const float* __restrict__ rowN,
    const v2f* __restrict__ colLo, const v2f* __restrict__ colHi,
    const float* __restrict__ colN,
    const float* __restrict__ gK,           // g_j * KC (column-indexed)
    const float* __restrict__ fpot,         // f (row potentials)
    int waveId, int strideWaves, int nStrips, int lane)
{
    const int half = lane >> 4;
    const int l    = lane & 15;
    const v2f* col = half ? colHi : colLo;
    float wacc = 0.0f;

    for (int sidx = 0; sidx < nStrips; ++sidx) {
        const int strip = (waveId + sidx * strideWaves) * 16;
        v2f a;
        if (half == 0) {
            v2f t = rowLo[strip + l];
            a.x = -2.0f * t.x; a.y = -2.0f * t.y;
        } else {
            v2f t = rowHi[strip + l];
            a.x = -2.0f * t.x; a.y = rowN[strip + l];
        }
        float fK[8];
#pragma unroll
        for (int k = 0; k < 8; ++k)
            fK[k] = fpot[strip + half * 8 + k] * KC;

        float acc[8];
#pragma unroll
        for (int k = 0; k < 8; ++k) acc[k] = 0.0f;
        v8f cz = {};

#pragma unroll 2
        for (int jt = 0; jt < NT; ++jt) {
            const int n   = jt * 16 + l;
            v2f   bb  = col[n];
            float gg  = gK[n];
            float cnj = colN[n];
            v8f d = __builtin_amdgcn_wmma_f32_16x16x4_f32(
                false, a, false, bb, (short)0, cz, false, false);
#pragma unroll
            for (int k = 0; k < 8; ++k) {
                float C = fmaxf(d[k] + cnj, 0.0f);
                float t2 = __builtin_fmaf(C, -KC, fK[k] + gg);
                acc[k] = __builtin_fmaf(__builtin_amdgcn_exp2f(t2), C, acc[k]);
            }
        }
        float p = 0.0f;
#pragma unroll
        for (int k = 0; k < 8; ++k) p += acc[k];
#pragma unroll
        for (int m = 1; m < 32; m <<= 1)
            p += __shfl_xor(p, m, 32);
        wacc += p;
    }
    return wacc;
}

// staging: batch slice of x,y -> formatted LDS operands
__device__ __forceinline__ void stage(
    const float* __restrict__ x, const float* __restrict__ y, int b,
    v2f* bXlo, v2f* bXhi, v2f* bYlo, v2f* bYhi, float* xn, float* yn,
    int tid, int nthreads)
{
    for (int i = tid; i < NPT; i += nthreads) {
        const float* xp = x + ((size_t)b * NPT + i) * 3;
        float x0 = xp[0], x1 = xp[1], x2 = xp[2];
        v2f t;
        t.x = x0; t.y = x1;   bXlo[i] = t;
        t.x = x2; t.y = 1.0f; bXhi[i] = t;
        xn[i] = x0 * x0 + x1 * x1 + x2 * x2;

        const float* yp = y + ((size_t)b * NPT + i) * 3;
        float y0 = yp[0], y1 = yp[1], y2 = yp[2];
        t.x = y0; t.y = y1;   bYlo[i] = t;
        t.x = y2; t.y = 1.0f; bYhi[i] = t;
        yn[i] = y0 * y0 + y1 * y1 + y2 * y2;
    }
}

// =======================================================================
// per-batch cross-workgroup barrier (sense-reversing, agent scope)
// =======================================================================
__device__ __forceinline__ void batch_barrier(unsigned* cnt, unsigned* gen)
{
    __threadfence();          // make this WG's stores device-visible
    __syncthreads();
    if (threadIdx.x == 0) {
        unsigned snap = __hip_atomic_load(gen, __ATOMIC_ACQUIRE,
                                          __HIP_MEMORY_SCOPE_AGENT);
        unsigned prev = __hip_atomic_fetch_add(cnt, 1u, __ATOMIC_ACQ_REL,
                                               __HIP_MEMORY_SCOPE_AGENT);
        if (prev == NWGB - 1) {
            __hip_atomic_store(cnt, 0u, __ATOMIC_RELAXED,
                               __HIP_MEMORY_SCOPE_AGENT);
            __hip_atomic_fetch_add(gen, 1u, __ATOMIC_RELEASE,
                                   __HIP_MEMORY_SCOPE_AGENT);
        } else {
            while (__hip_atomic_load(gen, __ATOMIC_ACQUIRE,
                                     __HIP_MEMORY_SCOPE_AGENT) == snap)
                __builtin_amdgcn_s_sleep(4);
        }
    }
    __syncthreads();
    __threadfence();          // invalidate stale lines before re-reading f/g
}

// =======================================================================
// init kernel: zero f,g, partials, barrier state (runs every launch so
// graph replays are deterministic; harness poisons ws once with 0xAA)
// =======================================================================
__global__ void init_ws_kernel(float* ws)
{
    const int i = blockIdx.x * blockDim.x + threadIdx.x;
    const int stride = gridDim.x * blockDim.x;
    for (int k = i; k < 2 * BATCH * NPT; k += stride) ws[k] = 0.0f;
    if (i < NWG) ws[P_OFF + i] = 0.0f;
    unsigned* bar = reinterpret_cast<unsigned*>(ws + BAR_OFF);
    if (i < BATCH * 64) bar[i] = 0u;
}

// =======================================================================
// main multi-workgroup kernel: 128 WGs x 256 threads
// =======================================================================
__global__ __launch_bounds__(TPB)
void sinkhorn_wmma_mwg_kernel(const float* __restrict__ x,
                              const float* __restrict__ y,
                              float* __restrict__ ws)
{
    __shared__ OperandBuf s;
    const int b    = blockIdx.x / NWGB;
    const int sub  = blockIdx.x % NWGB;
    const int tid  = threadIdx.x;
    const int wid  = tid >> 5;
    const int lane = tid & 31;
    const int bwave = sub * WAVES + wid;        // 0..63 within batch

    float* f_b = ws + F_OFF + (size_t)b * NPT;
    float* g_b = ws + G_OFF + (size_t)b * NPT;
    unsigned* bar = reinterpret_cast<unsigned*>(ws + BAR_OFF) + b * 64;
    unsigned* cnt = bar + 0;
    unsigned* gen = bar + 1;

    stage(x, y, b, s.bXlo, s.bXhi, s.bYlo, s.bYhi, s.xn, s.yn, tid, TPB);
    __syncthreads();

    for (int it = 0; it < ITERS; ++it) {
        // f update: needs current g (global, valid after init / prev barrier)
        for (int j = tid; j < NPT; j += TPB)
            s.aux[j] = (g_b[j] - s.yn[j]) * KC;
        __syncthreads();
        lse_pass(s.bXlo, s.bXhi, s.xn, s.bYlo, s.bYhi, s.aux,
                 f_b, bwave, BWAVES, SPW, lane);
        batch_barrier(cnt, gen);

        // g update: needs the full new f
        for (int i = tid; i < NPT; i += TPB)
            s.aux[i] = (f_b[i] - s.xn[i]) * KC;
        __syncthreads();
        lse_pass(s.bYlo, s.bYhi, s.yn, s.bXlo, s.bXhi, s.aux,
                 g_b, bwave, BWAVES, SPW, lane);
        batch_barrier(cnt, gen);
    }

    // final: sum_{i,j} P*C over this WG's strips
    for (int j = tid; j < NPT; j += TPB)
        s.aux[j] = g_b[j] * KC;
    __syncthreads();
    float wacc = dis_pass(s.bXlo, s.bXhi, s.xn, s.bYlo, s.bYhi, s.yn,
                          s.aux, f_b, bwave, BWAVES, SPW, lane);
    if (lane == 0) s.wsum[wid] = wacc;
    __syncthreads();
    if (tid == 0) {
        float t = 0.0f;
        for (int w = 0; w < WAVES; ++w) t += s.wsum[w];   // fixed order
        ws[P_OFF + blockIdx.x] = t;
    }
}

// =======================================================================
// fallback: self-contained, one 1024-thread WG per batch (needs only
// BATCH floats of workspace). Used when ws_size is too small.
// =======================================================================
#define FB_WAVES 32
#define FB_TPB   (FB_WAVES * 32)
#define FB_SPW   (NT / FB_WAVES)

__global__ __launch_bounds__(FB_TPB, 1)
void sinkhorn_wmma_fb_kernel(const float* __restrict__ x,
                             const float* __restrict__ y,
                             float* __restrict__ bsum)
{
    __shared__ OperandBufFB s;
    const int b    = blockIdx.x;
    const int tid  = threadIdx.x;
    const int wid  = tid >> 5;
    const int lane = tid & 31;

    stage(x, y, b, s.bXlo, s.bXhi, s.bYlo, s.bYhi, s.xn, s.yn, tid, FB_TPB);
    for (int i = tid; i < NPT; i += FB_TPB) { s.f[i] = 0.0f; s.g[i] = 0.0f; }
    __syncthreads();

    for (int it = 0; it < ITERS; ++it) {
        for (int j = tid; j < NPT; j += FB_TPB)
            s.aux[j] = (s.g[j] - s.yn[j]) * KC;
        __syncthreads();
        lse_pass(s.bXlo, s.bXhi, s.xn, s.bYlo, s.bYhi, s.aux,
                 s.f, wid, FB_WAVES, FB_SPW, lane);
        __syncthreads();
        for (int i = tid; i < NPT; i += FB_TPB)
            s.aux[i] = (s.f[i] - s.xn[i]) * KC;
        __syncthreads();
        lse_pass(s.bYlo, s.bYhi, s.yn, s.bXlo, s.bXhi, s.aux,
                 s.g, wid, FB_WAVES, FB_SPW, lane);
        __syncthreads();
    }

    for (int j = tid; j < NPT; j += FB_TPB)
        s.aux[j] = s.g[j] * KC;
    __syncthreads();
    float wacc = dis_pass(s.bXlo, s.bXhi, s.xn, s.bYlo, s.bYhi, s.yn,
                          s.aux, s.f, wid, FB_WAVES, FB_SPW, lane);
    if (lane == 0) s.wsum[wid] = wacc;
    __syncthreads();
    if (tid == 0) {
        float t = 0.0f;
        for (int w = 0; w < FB_WAVES; ++w) t += s.wsum[w];
        bsum[b] = t;
    }
}

__global__ void finalize_kernel(const float* __restrict__ partial, int n,
                                float* __restrict__ out)
{
    if (threadIdx.x == 0 && blockIdx.x == 0) {
        float t = 0.0f;
        for (int i = 0; i < n; ++i) t += partial[i];    // fixed order
        // dis.mean() = sum_{b,i} rowsum * N / (B*N) = sum(rowsum)/B
        out[0] = t * (1.0f / (float)BATCH);
    }
}

extern "C" void kernel_launch(void* const* d_in, const int* in_sizes, int n_in,
                              void* d_out, int out_size, void* d_ws, size_t ws_size,
                              hipStream_t stream)
{
    const float* x = (const float*)d_in[0];
    const float* y = (const float*)d_in[1];
    float* out = (float*)d_out;
    float* ws  = (float*)d_ws;

    if (ws_size >= WS_BYTES_NEEDED) {
        init_ws_kernel<<<128, 256, 0, stream>>>(ws);
        sinkhorn_wmma_mwg_kernel<<<NWG, TPB, 0, stream>>>(x, y, ws);
        finalize_kernel<<<1, 32, 0, stream>>>(ws + P_OFF, NWG, out);
    } else {
        // minimal-workspace fallback: 16 self-contained workgroups
        sinkhorn_wmma_fb_kernel<<<BATCH, FB_TPB, 0, stream>>>(x, y, ws);
        finalize_kernel<<<1, 32, 0, stream>>>(ws, BATCH, out);
    }
}